// ChamferLoss_35502199669085
// MI455X (gfx1250) — compile-verified
//
#include <hip/hip_runtime.h>

typedef __attribute__((ext_vector_type(2))) float v2f;
typedef __attribute__((ext_vector_type(8))) float v8f;

#define TILE_M 2048   // packed y-points staged in LDS per super-chunk (32 KB)

// Pack (p0,p1,p2) -> (-2p0, -2p1, -2p2, ||p||^2) so the WMMA A-operand build
// in the hot loop is a single 8-byte load.
__global__ __launch_bounds__(256) void chamfer_pack_kernel(
    const float* __restrict__ pts, float4* __restrict__ packed, int total)
{
    int i = blockIdx.x * blockDim.x + threadIdx.x;
    if (i < total) {
        float a = pts[3 * i + 0];
        float b = pts[3 * i + 1];
        float c = pts[3 * i + 2];
        float4 v;
        v.x = -2.0f * a;
        v.y = -2.0f * b;
        v.z = -2.0f * c;
        v.w = a * a + b * b + c * c;
        packed[i] = v;
    }
}

// For each "column" point x_n (kept), compute min over all "row" points y_m of
// ||x_n - y_m||^2, using V_WMMA_F32_16X16X4_F32 tiles:
//   A row m  = [-2y0, -2y1, -2y2, ||y||^2]   (from packed Y)
//   B col n  = [x0, x1, x2, 1]
//   D[m][n]  = -2 x.y + ||y||^2   -> min over m, then add ||x||^2 at the end.
// Block = 256 threads = 8 waves; each wave owns 16 columns. Y tiles are shared
// by all waves, staged through LDS in 2048-point super-chunks.
__global__ __launch_bounds__(256) void chamfer_min_kernel(
    const float*  __restrict__ X,     // [B, Ncols, 3] raw column points
    const float4* __restrict__ Ypk,   // [B, Mrows] packed row points
    float*        __restrict__ outmin,// [B, Ncols]
    int Ncols, int Mrows)
{
    __shared__ float4 tile[TILE_M];

    const int lane = threadIdx.x & 31;
    const int wave = threadIdx.x >> 5;
    const int tilesPerBatch = Ncols >> 7;           // 128 columns per block
    const int b  = blockIdx.x / tilesPerBatch;
    const int ct = blockIdx.x % tilesPerBatch;
    const int n  = (ct << 7) + (wave << 4) + (lane & 15);
    const bool hi = (lane >= 16);
    const int  sel = hi ? 1 : 0;                    // lo half: K=0,1  hi half: K=2,3

    const float*  Xb = X + ((size_t)b * Ncols + n) * 3;
    const float4* Yb = Ypk + (size_t)b * Mrows;

    const float x0 = Xb[0], x1 = Xb[1], x2 = Xb[2];
    const float xnorm = x0 * x0 + x1 * x1 + x2 * x2;

    // B-operand (4x16): lanes 0-15 hold K=0,1 ; lanes 16-31 hold K=2,3
    v2f bm;
    bm[0] = hi ? x2 : x0;
    bm[1] = hi ? 1.0f : x1;

    float runmin = 3.4e38f;

    for (int mc = 0; mc < Mrows; mc += TILE_M) {
        __syncthreads();
        for (int i = threadIdx.x; i < TILE_M; i += 256)
            tile[i] = Yb[mc + i];
        __syncthreads();

#pragma unroll 4
        for (int m0 = 0; m0 < TILE_M; m0 += 16) {
            // A-operand (16x4): one ds_load_b64 per lane
            const float2* p = reinterpret_cast<const float2*>(&tile[m0 + (lane & 15)]);
            float2 av = p[sel];
            v2f am;
            am[0] = av.x;
            am[1] = av.y;

            v8f c = {};
            v8f d = __builtin_amdgcn_wmma_f32_16x16x4_f32(
                false, am, false, bm, (short)0, c, false, false);

            // min over rows m0..m0+7 (this lane-half) of this tile
            float m01 = fminf(d[0], d[1]);
            float m23 = fminf(d[2], d[3]);
            float m45 = fminf(d[4], d[5]);
            float m67 = fminf(d[6], d[7]);
            float m   = fminf(fminf(m01, m23), fminf(m45, m67));
            runmin = fminf(runmin, m);
        }
    }

    // combine the two lane-halves (rows 0-7 vs 8-15 of every tile)
    runmin = fminf(runmin, __shfl_xor(runmin, 16, 32));

    if (!hi)
        outmin[(size_t)b * Ncols + n] = runmin + xnorm;
}

__global__ __launch_bounds__(256) void chamfer_reduce_kernel(
    const float* __restrict__ min1, const float* __restrict__ min2,
    int n1, int n2, float* __restrict__ out)
{
    __shared__ float s1s[256];
    __shared__ float s2s[256];
    const int tid = threadIdx.x;

    float s1 = 0.0f, s2 = 0.0f;
    for (int i = tid; i < n1; i += 256) s1 += fmaxf(min1[i], 1e-10f);
    for (int i = tid; i < n2; i += 256) s2 += fmaxf(min2[i], 1e-10f);
    s1s[tid] = s1;
    s2s[tid] = s2;
    __syncthreads();
    for (int off = 128; off > 0; off >>= 1) {
        if (tid < off) {
            s1s[tid] += s1s[tid + off];
            s2s[tid] += s2s[tid + off];
        }
        __syncthreads();
    }
    if (tid == 0)
        out[0] = (s1s[0] / (float)n1 + s2s[0] / (float)n2) * 1000.0f;
}

extern "C" void kernel_launch(void* const* d_in, const int* in_sizes, int n_in,
                              void* d_out, int out_size, void* d_ws, size_t ws_size,
                              hipStream_t stream) {
    const float* gt  = (const float*)d_in[0];  // ground_truth  [B,N,3]
    const float* rec = (const float*)d_in[1];  // reconstruction [B,M,3]

    const int B = 4, D = 3;
    const int N = in_sizes[0] / (B * D);       // 8192
    const int M = in_sizes[1] / (B * D);       // 8192

    float*  ws     = (float*)d_ws;
    float4* gt_pk  = (float4*)ws;                          // B*N float4
    float4* rec_pk = (float4*)(ws + 4 * (size_t)B * N);    // B*M float4
    float*  min1   = ws + 4 * (size_t)B * N + 4 * (size_t)B * M;  // B*M floats
    float*  min2   = min1 + (size_t)B * M;                        // B*N floats

    const int totG = B * N, totR = B * M;
    chamfer_pack_kernel<<<(totG + 255) / 256, 256, 0, stream>>>(gt,  gt_pk,  totG);
    chamfer_pack_kernel<<<(totR + 255) / 256, 256, 0, stream>>>(rec, rec_pk, totR);

    // loss_1: for each reconstruction point, min over ground-truth points
    chamfer_min_kernel<<<B * (M / 128), 256, 0, stream>>>(rec, gt_pk,  min1, M, N);
    // loss_2: for each ground-truth point, min over reconstruction points
    chamfer_min_kernel<<<B * (N / 128), 256, 0, stream>>>(gt,  rec_pk, min2, N, M);

    chamfer_reduce_kernel<<<1, 256, 0, stream>>>(min1, min2, B * M, B * N,
                                                 (float*)d_out);
}